// CustomLoss_77653008712354
// MI455X (gfx1250) — compile-verified
//
#include <hip/hip_runtime.h>
#include <hip/hip_bf16.h>

#define NCLS 38
#define NPAIR 19                  // NCLS/2, pairs never cross a row block

typedef __attribute__((ext_vector_type(2))) float v2f;
typedef __attribute__((ext_vector_type(8))) float v8f;
typedef __attribute__((ext_vector_type(2))) int   v2i;

// Exact 32-lane sum using one V_WMMA_F32_16X16X4_F32.
// A: lane L<16 -> A[L][0]=v ; lane L>=16 -> A[L-16][2]=v ; other K slots 0.
// B: all ones (4x16) -> D[m][n] = v_m + v_{m+16}.
// Lane n sums its 8 D rows (M=0..7 or 8..15), xor-16 combine -> full total.
__device__ __forceinline__ float wave_sum32(float v) {
    v2f a; a.x = v;    a.y = 0.0f;
    v2f b; b.x = 1.0f; b.y = 1.0f;
    v8f c = {};
    c = __builtin_amdgcn_wmma_f32_16x16x4_f32(
        false, a, false, b, (short)0, c, false, false);
    float s = c[0] + c[1] + c[2] + c[3] + c[4] + c[5] + c[6] + c[7];
    s += __shfl_xor(s, 16, 32);
    return s;
}

__device__ __forceinline__ float block_sum(float v) {
    __shared__ float wsum[8];              // 256 threads = 8 wave32
    float s = wave_sum32(v);
    const int wave = threadIdx.x >> 5;
    const int lane = threadIdx.x & 31;
    if (lane == 0) wsum[wave] = s;
    __syncthreads();
    float tot = 0.0f;
    if (threadIdx.x == 0) {
        const int nw = blockDim.x >> 5;
        for (int w = 0; w < nw; ++w) tot += wsum[w];
    }
    return tot;                            // valid in thread 0
}

// Base-2 fused loss element (a = logit * log2e, y in {0,1}):
//   t2  = log2(1 + exp2(-|a|))            (softplus in bits)
//   bce2   = max(a,0) - (y?a:0) + t2
//   l2p    = min(a,0) - t2 ;  l2p1 = -max(a,0) - t2   (log2-sigmoids)
//   kl2    = q*(l2q-l2p) + q1*(l2q1-l2p1),  q = exp2(l2q), q1 = exp2(l2q1)
//   elem   = ln2 * (bce2 + 0.5*kl2)       (ln2 applied once, at the end)
__global__ void CustomLoss_partial_kernel(const float* __restrict__ logits,
                                          const int*   __restrict__ labels,
                                          float*       __restrict__ partial,
                                          unsigned int nPairs) {
    const float K = 1.44269504088896340736f;   // log2(e)
    unsigned int p      = blockIdx.x * blockDim.x + threadIdx.x;
    unsigned int stride = gridDim.x * blockDim.x;
    float acc = 0.0f;

    for (; p < nPairs; p += stride) {
        unsigned int b  = p / NPAIR;           // magic-multiply
        unsigned int jp = p - b * NPAIR;
        const float* row = logits + (size_t)b * (2 * NCLS);

        v2f slv = __builtin_nontemporal_load((const v2f*)row + jp);
        v2f rlv = __builtin_nontemporal_load((const v2f*)(row + NCLS) + jp);
        v2i yv  = __builtin_nontemporal_load((const v2i*)labels + p);

#pragma unroll
        for (int e = 0; e < 2; ++e) {
            float as  = slv[e] * K;
            float ts  = __builtin_amdgcn_logf(1.0f +
                        __builtin_amdgcn_exp2f(-fabsf(as)));   // log2(1+2^-|as|)
            float mxs = fmaxf(as, 0.0f);
            float sy  = yv[e] ? as : 0.0f;                     // v_cndmask
            float bce2 = mxs - sy + ts;
            float l2p  = fminf(as, 0.0f) - ts;
            float l2p1 = -mxs - ts;

            float ar  = rlv[e] * K;
            float u   = __builtin_amdgcn_logf(1.0f +
                        __builtin_amdgcn_exp2f(-fabsf(ar)));
            float mxr = fmaxf(ar, 0.0f);
            float l2q  = fminf(ar, 0.0f) - u;
            float l2q1 = -mxr - u;
            float q   = __builtin_amdgcn_exp2f(l2q);
            float q1  = __builtin_amdgcn_exp2f(l2q1);
            float kl2 = q * (l2q - l2p) + q1 * (l2q1 - l2p1);

            acc += bce2 + 0.5f * kl2;
        }
    }
    float tot = block_sum(acc);                // all lanes reconverged
    if (threadIdx.x == 0) partial[blockIdx.x] = tot;
}

__global__ void CustomLoss_final_kernel(const float* __restrict__ partial,
                                        int nPart,
                                        float* __restrict__ out,
                                        float scale) {   // ln2 / N
    float acc = 0.0f;
    for (int i = threadIdx.x; i < nPart; i += blockDim.x)
        acc += partial[i];
    float tot = block_sum(acc);
    if (threadIdx.x == 0) out[0] = tot * scale;
}

extern "C" void kernel_launch(void* const* d_in, const int* in_sizes, int n_in,
                              void* d_out, int out_size, void* d_ws, size_t ws_size,
                              hipStream_t stream) {
    const float* logits = (const float*)d_in[0];   // (B, 76) f32
    const int*   labels = (const int*)d_in[1];     // (B, 38) i32
    (void)n_in; (void)out_size; (void)ws_size;

    const unsigned int n      = (unsigned int)in_sizes[1]; // B*38
    const unsigned int nPairs = n / 2u;                    // B*19, exact
    float* partial = (float*)d_ws;

    const int NB = 2048;     // 2048 block partials (8 KB in d_ws)
    const int BT = 256;      // 8 wave32 per block

    const float LN2 = 0.69314718055994530942f;

    CustomLoss_partial_kernel<<<NB, BT, 0, stream>>>(logits, labels, partial, nPairs);
    CustomLoss_final_kernel<<<1, BT, 0, stream>>>(partial, NB, (float*)d_out,
                                                  LN2 / (float)n);
}